// Network_83141976916791
// MI455X (gfx1250) — compile-verified
//
#include <hip/hip_runtime.h>

typedef __attribute__((ext_vector_type(2))) float v2f;
typedef __attribute__((ext_vector_type(8))) float v8f;

#define TPB 256

// Native CDNA5 float atomic-add (no return -> STOREcnt; kernel-end waits).
__device__ __forceinline__ void atomic_add_f32(float* p, float v) {
    asm volatile("global_atomic_add_f32 %0, %1, off"
                 :
                 : "v"(p), "v"(v)
                 : "memory");
}

// ---------------- degree / normalization ----------------

__global__ void k_init_deg(float* deg, int n) {
    int i = blockIdx.x * blockDim.x + threadIdx.x;
    if (i < n) deg[i] = 1.0f;  // self-loop
}

__global__ void k_deg_scatter(const int* dst, float* deg, int E) {
    int e = blockIdx.x * blockDim.x + threadIdx.x;
    if (e < E) atomic_add_f32(&deg[dst[e]], 1.0f);
}

__global__ void k_dinv(float* dinv, float* selfn, int n) {
    int i = blockIdx.x * blockDim.x + threadIdx.x;
    if (i < n) {
        float d = dinv[i];
        float r = (d > 0.0f) ? rsqrtf(d) : 0.0f;
        dinv[i]  = r;
        selfn[i] = r * r;
    }
}

__global__ void k_norm(const int* src, const int* dst, const float* dinv,
                       float* norm, int E) {
    int e = blockIdx.x * blockDim.x + threadIdx.x;
    if (e < E) norm[e] = dinv[src[e]] * dinv[dst[e]];
}

// ---------------- weight zero-padding ----------------

__global__ void k_pad_w(const float* in, float* out, int rin, int cin,
                        int rout, int cout) {
    int i = blockIdx.x * blockDim.x + threadIdx.x;
    if (i >= rout * cout) return;
    int r = i / cout, c = i % cout;
    out[i] = (r < rin && c < cin) ? in[r * cin + c] : 0.0f;
}

// ---------------- fp32 WMMA GEMM: C[N x dpad] = A[N x K] * B[K x dpad] ----

__global__ void k_gemm_wmma(const float* __restrict__ A,
                            const float* __restrict__ B,
                            float* __restrict__ C,
                            int K, int lda, int ldb, int ldc,
                            int ncolt, int ntiles) {
    int wid  = (blockIdx.x * blockDim.x + threadIdx.x) >> 5;
    int lane = threadIdx.x & 31;
    if (wid >= ntiles) return;
    int tn = wid % ncolt;
    int tm = wid / ncolt;
    int row  = lane & 15;
    int half = lane >> 4;

    const float* arow = A + (tm * 16 + row) * lda;
    const float* bcol = B + tn * 16 + row;

    v8f acc = {};
    for (int kb = 0; kb < K; kb += 4) {
        int kk = kb + half * 2;
        v2f a, b;
        a.x = arow[kk];
        a.y = arow[kk + 1];
        b.x = bcol[kk * ldb];
        b.y = bcol[(kk + 1) * ldb];
        acc = __builtin_amdgcn_wmma_f32_16x16x4_f32(
            false, a, false, b, (short)0, acc, false, false);
    }

    float* cbase = C + tn * 16 + row;
#pragma unroll
    for (int v = 0; v < 8; ++v) {
        int rm = v + half * 8;
        cbase[(tm * 16 + rm) * ldc] = acc[v];
    }
}

// ---------------- output init: bias + self-loop contribution ----------------

__global__ void k_init_out(const float* __restrict__ XW,
                           const float* __restrict__ selfn,
                           const float* __restrict__ bias, int bn,
                           float* __restrict__ out, int total, int sh) {
    int i = blockIdx.x * blockDim.x + threadIdx.x;
    if (i >= total) return;
    int node = i >> sh;
    int k = i & ((1 << sh) - 1);
    float b = (k < bn) ? bias[k] : 0.0f;
    out[i] = b + selfn[node] * XW[i];
}

// ---------------- edge scatter, d = 64 (one wave per edge) -----------------
// Lane k handles columns {2k, 2k+1}: one coalesced b64 gather per lane,
// two adjacent-address native f32 atomics (same 128B sector).

__global__ void k_scatter64(const int* __restrict__ src,
                            const int* __restrict__ dst,
                            const float* __restrict__ norm,
                            const float* __restrict__ XW,
                            float* __restrict__ out, int E) {
    int t = blockIdx.x * blockDim.x + threadIdx.x;
    int e = t >> 5;
    int lane = t & 31;
    if (e >= E) return;
    int s = src[e], d = dst[e];
    float nv = norm[e];
    const float2 xv = *(const float2*)(XW + s * 64 + lane * 2);
    float* op = out + d * 64 + lane * 2;
    atomic_add_f32(op,     nv * xv.x);
    atomic_add_f32(op + 1, nv * xv.y);
}

// ---------------- edge scatter, padded d = 16 with dreal valid cols --------

__global__ void k_scatter16(const int* __restrict__ src,
                            const int* __restrict__ dst,
                            const float* __restrict__ norm,
                            const float* __restrict__ XW,
                            float* __restrict__ out, int E, int dreal) {
    int t = blockIdx.x * blockDim.x + threadIdx.x;
    int e = t >> 4;
    int col = t & 15;
    if (e >= E || col >= dreal) return;
    atomic_add_f32(&out[dst[e] * 16 + col], norm[e] * XW[src[e] * 16 + col]);
}

// ---------------- ELU ----------------

__global__ void k_elu(float* h, int total) {
    int i = blockIdx.x * blockDim.x + threadIdx.x;
    if (i >= total) return;
    float x = h[i];
    h[i] = (x > 0.0f) ? x : (__expf(x) - 1.0f);
}

// ---------------- classifier: out[i] = H[i,0:11] . Wc + bc ----------------

__global__ void k_classifier(const float* __restrict__ H,
                             const float* __restrict__ Wc,
                             const float* __restrict__ bc,
                             float* __restrict__ out, int n) {
    int i = blockIdx.x * blockDim.x + threadIdx.x;
    if (i >= n) return;
    float s = bc[0];
    const float* hr = H + i * 16;
#pragma unroll
    for (int k = 0; k < 11; ++k) s += hr[k] * Wc[k];
    out[i] = s;
}

// ---------------- host orchestration ----------------

static inline int cdiv(long long a, int b) { return (int)((a + b - 1) / b); }

static void run_layer(const float* Hin, int K, int lda,
                      const float* W, int ldb,
                      const float* bias, int bn,
                      int dpad, int dreal,
                      const int* src, const int* dst, const float* norm,
                      const float* selfn, float* XW, float* out,
                      int N, int E, hipStream_t stream) {
    int ncolt  = dpad / 16;
    int ntiles = (N / 16) * ncolt;
    long long gthreads = (long long)ntiles * 32;
    k_gemm_wmma<<<cdiv(gthreads, TPB), TPB, 0, stream>>>(
        Hin, W, XW, K, lda, ldb, dpad, ncolt, ntiles);

    int total = N * dpad;
    int sh = (dpad == 64) ? 6 : 4;
    k_init_out<<<cdiv(total, TPB), TPB, 0, stream>>>(
        XW, selfn, bias, bn, out, total, sh);

    if (dpad == 64) {
        long long st = (long long)E * 32;
        k_scatter64<<<cdiv(st, TPB), TPB, 0, stream>>>(
            src, dst, norm, XW, out, E);
    } else {
        long long st = (long long)E * 16;
        k_scatter16<<<cdiv(st, TPB), TPB, 0, stream>>>(
            src, dst, norm, XW, out, E, dreal);
    }

    k_elu<<<cdiv(total, TPB), TPB, 0, stream>>>(out, total);
}

extern "C" void kernel_launch(void* const* d_in, const int* in_sizes, int n_in,
                              void* d_out, int out_size, void* d_ws, size_t ws_size,
                              hipStream_t stream) {
    const float* x    = (const float*)d_in[0];
    const int*   eidx = (const int*)d_in[1];
    const float* W1 = (const float*)d_in[2];  const float* b1 = (const float*)d_in[3];
    const float* W2 = (const float*)d_in[4];  const float* b2 = (const float*)d_in[5];
    const float* W3 = (const float*)d_in[6];  const float* b3 = (const float*)d_in[7];
    const float* W4 = (const float*)d_in[8];  const float* b4 = (const float*)d_in[9];
    const float* W5 = (const float*)d_in[10]; const float* b5 = (const float*)d_in[11];
    const float* Wc = (const float*)d_in[12]; const float* bc = (const float*)d_in[13];

    const int N = in_sizes[0] / 64;
    const int E = in_sizes[1] / 2;
    const int* src = eidx;
    const int* dst = eidx + E;

    // workspace carving (256B aligned)
    char* ws = (char*)d_ws;
    size_t off = 0;
    auto carve = [&](size_t bytes) -> float* {
        float* p = (float*)(ws + off);
        off = (off + bytes + 255) & ~(size_t)255;
        return p;
    };
    float* norm  = carve((size_t)E * 4);
    float* dinv  = carve((size_t)N * 4);   // holds deg first, then dinv
    float* selfn = carve((size_t)N * 4);
    float* W4p   = carve(64 * 16 * 4);
    float* W5p   = carve(16 * 16 * 4);
    float* XW    = carve((size_t)N * 64 * 4);
    float* H1    = carve((size_t)N * 64 * 4);
    float* H2    = carve((size_t)N * 64 * 4);

    // normalization: deg -> dinv -> per-edge norm
    k_init_deg<<<cdiv(N, TPB), TPB, 0, stream>>>(dinv, N);
    k_deg_scatter<<<cdiv(E, TPB), TPB, 0, stream>>>(dst, dinv, E);
    k_dinv<<<cdiv(N, TPB), TPB, 0, stream>>>(dinv, selfn, N);
    k_norm<<<cdiv(E, TPB), TPB, 0, stream>>>(src, dst, dinv, norm, E);

    // zero-pad W4 (64x11 -> 64x16) and W5 (11x11 -> 16x16)
    k_pad_w<<<cdiv(64 * 16, TPB), TPB, 0, stream>>>(W4, W4p, 64, 11, 64, 16);
    k_pad_w<<<cdiv(16 * 16, TPB), TPB, 0, stream>>>(W5, W5p, 11, 11, 16, 16);

    // 5 GCN layers
    run_layer(x,  64, 64, W1,  64, b1, 64, 64, 64, src, dst, norm, selfn, XW, H1, N, E, stream);
    run_layer(H1, 64, 64, W2,  64, b2, 64, 64, 64, src, dst, norm, selfn, XW, H2, N, E, stream);
    run_layer(H2, 64, 64, W3,  64, b3, 64, 64, 64, src, dst, norm, selfn, XW, H1, N, E, stream);
    run_layer(H1, 64, 64, W4p, 16, b4, 11, 16, 11, src, dst, norm, selfn, XW, H2, N, E, stream);
    run_layer(H2, 16, 16, W5p, 16, b5, 11, 16, 11, src, dst, norm, selfn, XW, H1, N, E, stream);

    // classifier
    k_classifier<<<cdiv(N, TPB), TPB, 0, stream>>>(H1, Wc, bc, (float*)d_out, N);
}